// ChGKModel_79903571574852
// MI455X (gfx1250) — compile-verified
//
#include <hip/hip_runtime.h>

#define EPS 1e-7f
#define NUM_TYPES 3
#define MAX_TS 10
#define TILE_ELEMS 2048            // 256 threads * 8 elements

typedef unsigned int u32x4 __attribute__((ext_vector_type(4)));
typedef int          i32x8 __attribute__((ext_vector_type(8)));
typedef int          i32x4 __attribute__((ext_vector_type(4)));

// ---------------------------------------------------------------------------
// TDM helper: 1-D DMA of tile_elems 4-byte elements from global to LDS.
// Descriptor per CDNA5 ISA D# layout (CDNA5 ISA):
//   group0: [1:0]=count=1, [63:32]=lds_addr, [120:64]=global_addr, [127:126]=2
//   group1: data_size=2 (4B) at [17:16]; tensor_dim0 at [79:48] (OOB -> zero);
//           tile_dim0 at [127:112]; tile_dim1=0 (1-D); strides unused.
// This toolchain exposes the 6-arg builtin:
//   (u32x4 g0, i32x8 g1, i32x4 g2, i32x4 g3, i32x8 g4, i32 cpol)
// ---------------------------------------------------------------------------
#if __has_builtin(__builtin_amdgcn_tensor_load_to_lds)
#define HAVE_TDM 1
__device__ __forceinline__ void tdm_load_1d(unsigned lds_off_bytes,
                                            const void* gptr,
                                            unsigned tile_elems,
                                            unsigned long long tensor_elems) {
    unsigned long long ga = (unsigned long long)(size_t)gptr;
    unsigned td0 = (tensor_elems > 0xFFFFFFFFull) ? 0xFFFFFFFFu
                                                  : (unsigned)tensor_elems;
    u32x4 g0;
    g0.x = 0x1u;                                          // count=1, user mode
    g0.y = lds_off_bytes;                                 // lds_addr (bytes)
    g0.z = (unsigned)(ga & 0xFFFFFFFFu);                  // global_addr[31:0]
    g0.w = (unsigned)((ga >> 32) & 0x01FFFFFFu)           // global_addr[56:32]
         | (2u << 30);                                    // type = 2 ("image")
    i32x8 g1;
    g1[0] = (int)(2u << 16);                              // data_size=4B
    g1[1] = (int)((td0 & 0xFFFFu) << 16);                 // tensor_dim0[15:0]
    g1[2] = (int)((td0 >> 16) & 0xFFFFu);                 // tensor_dim0[31:16]
    g1[3] = (int)((tile_elems & 0xFFFFu) << 16);          // tile_dim0
    g1[4] = 0;                                            // tile_dim1/2 unused
    g1[5] = 0;                                            // dim0 stride unused
    g1[6] = 0;
    g1[7] = 0;
    i32x4 z4 = {0, 0, 0, 0};                              // groups 2/3 unused
    i32x8 z8 = {0, 0, 0, 0, 0, 0, 0, 0};                  // trailing group unused
    __builtin_amdgcn_tensor_load_to_lds(g0, g1, z4, z4, z8, 0);
}
#else
#define HAVE_TDM 0
#endif

// ---------------------------------------------------------------------------
// Kernel 1: per-type count / sum / sum-of-squares over tournament_dl.
// wave32 shuffle reduction -> LDS cross-wave reduction -> 9 atomics per block.
// ---------------------------------------------------------------------------
__global__ void __launch_bounds__(256)
k_dl_stats(const float* __restrict__ dl, const int* __restrict__ ttype,
           int n, float* __restrict__ stats /* [9]: cnt[3], sum[3], sumsq[3] */) {
    float acc[9];
#pragma unroll
    for (int i = 0; i < 9; ++i) acc[i] = 0.0f;

    int idx    = blockIdx.x * blockDim.x + threadIdx.x;
    int stride = gridDim.x * blockDim.x;
    for (int i = idx; i < n; i += stride) {
        float v = dl[i];
        int   t = ttype[i];
#pragma unroll
        for (int t2 = 0; t2 < NUM_TYPES; ++t2) {
            float m = (t == t2) ? 1.0f : 0.0f;   // branchless per-type select
            acc[t2]     += m;
            acc[3 + t2] += m * v;
            acc[6 + t2] += m * v * v;
        }
    }
#pragma unroll
    for (int i = 0; i < 9; ++i) {
#pragma unroll
        for (int off = 16; off > 0; off >>= 1)
            acc[i] += __shfl_down(acc[i], off, 32);
    }
    __shared__ float sm[9][8];
    int lane = threadIdx.x & 31;
    int wv   = threadIdx.x >> 5;
    if (lane == 0) {
#pragma unroll
        for (int i = 0; i < 9; ++i) sm[i][wv] = acc[i];
    }
    __syncthreads();
    if (threadIdx.x == 0) {
        int nw = blockDim.x >> 5;
#pragma unroll
        for (int i = 0; i < 9; ++i) {
            float s = 0.0f;
            for (int w = 0; w < nw; ++w) s += sm[i][w];
            atomicAdd(&stats[i], s);
        }
    }
}

// ---------------------------------------------------------------------------
// Kernel 2: fold stats + scale/bias into per-type affine coeffs:
//   b_i = b[q] + k0[t] + k1[t] * dl[q],  k1 = scale/std, k0 = bias - k1*mean
// ---------------------------------------------------------------------------
__global__ void k_finalize_stats(const float* __restrict__ stats,
                                 const float* __restrict__ dl_scale,
                                 const float* __restrict__ type_bias,
                                 float* __restrict__ k01 /* [6] */) {
    int t = threadIdx.x;
    if (t < NUM_TYPES) {
        float c  = stats[t];
        float sc = fmaxf(c, 1.0f);
        float mean = stats[3 + t] / sc;
        float ex2  = stats[6 + t] / sc;
        float var  = fmaxf(ex2 - mean * mean, 0.0f);
        float sd   = sqrtf(var);
        mean = (c > 0.0f) ? mean : 0.0f;
        sd   = ((c > 0.0f) && (sd > 1e-6f)) ? sd : 1.0f;
        float k1 = dl_scale[t] / sd;
        float k0 = type_bias[t] - k1 * mean;
        k01[t]     = k0;
        k01[3 + t] = k1;
    }
}

// ---------------------------------------------------------------------------
// Kernel 3: per-batch-question a_i, b_i. Gather targets are L2-resident.
// ---------------------------------------------------------------------------
__global__ void __launch_bounds__(256)
k_question(const int* __restrict__ qidx, const int* __restrict__ ttype,
           const float* __restrict__ dl, const float* __restrict__ bq,
           const float* __restrict__ log_a, const float* __restrict__ k01,
           float* __restrict__ a_i, float* __restrict__ b_i, int Bn) {
    __shared__ float s_k0[NUM_TYPES], s_k1[NUM_TYPES];
    if (threadIdx.x < NUM_TYPES) {
        s_k0[threadIdx.x] = k01[threadIdx.x];
        s_k1[threadIdx.x] = k01[NUM_TYPES + threadIdx.x];
    }
    __syncthreads();

    int i      = blockIdx.x * blockDim.x + threadIdx.x;
    int stride = gridDim.x * blockDim.x;
    for (; i < Bn; i += stride) {
        __builtin_prefetch(qidx + i + stride, 0, 1);   // global_prefetch_b8
        int q = qidx[i];
        int t = ttype[q];
        b_i[i] = bq[q] + s_k0[t] + s_k1[t] * dl[q];
        a_i[i] = fmaxf(__expf(log_a[q]), EPS);
    }
}

// ---------------------------------------------------------------------------
// Shared compute body for one 8-element chunk (sorted-segment run accumulate).
// ---------------------------------------------------------------------------
__device__ __forceinline__ void accum_chunk(const int* sg, const float* th, int m,
                                            const float* __restrict__ a_i,
                                            const float* __restrict__ b_i,
                                            float* __restrict__ lam,
                                            int* __restrict__ cnt) {
    int   cur = sg[0];
    float a   = a_i[cur];
    float bb  = b_i[cur];
    float acc = 0.0f;
    int   c   = 0;
#pragma unroll
    for (int j = 0; j < 8; ++j) {
        if (j >= m) break;
        int s = sg[j];
        if (s != cur) {
            atomicAdd(&lam[cur], acc);
            atomicAdd(&cnt[cur], c);
            cur = s; acc = 0.0f; c = 0;
            a = a_i[s]; bb = b_i[s];
        }
        float logit = fminf(fmaxf(a * th[j] - bb, -20.0f), 20.0f);
        acc += __expf(logit);
        ++c;
    }
    atomicAdd(&lam[cur], acc);
    atomicAdd(&cnt[cur], c);
}

// ---------------------------------------------------------------------------
// Kernel 4: hot loop. TDM double-buffers the two contiguous index streams
// into LDS (tile t+1 DMA overlaps tile t compute; TENSORcnt tracks it), the
// vector pipe is reserved for the random theta gathers (L2-resident), and
// segmented sums hit lam/cnt with ~2.3 atomics per 8-chunk.
// ---------------------------------------------------------------------------
#if HAVE_TDM
__global__ void __launch_bounds__(256)
k_accum(const int* __restrict__ seg, const int* __restrict__ pidx,
        const float* __restrict__ theta,
        const float* __restrict__ a_i, const float* __restrict__ b_i,
        float* __restrict__ lam, int* __restrict__ cnt, int total) {
    __shared__ int sseg[2][TILE_ELEMS];
    __shared__ int spid[2][TILE_ELEMS];

    int numTiles = (total + TILE_ELEMS - 1) / TILE_ELEMS;
    int tile     = blockIdx.x;
    if (tile >= numTiles) return;

    const bool issuer = (threadIdx.x < 32);   // wave 0 drives the TDM

    if (issuer) {                             // prologue: tile -> buffer 0
        int start = tile * TILE_ELEMS;
        unsigned long long rem = (unsigned long long)(total - start);
        tdm_load_1d((unsigned)(size_t)&sseg[0][0], seg + start, TILE_ELEMS, rem);
        tdm_load_1d((unsigned)(size_t)&spid[0][0], pidx + start, TILE_ELEMS, rem);
    }

    int cur = 0;
    for (; tile < numTiles; tile += gridDim.x) {
        int nextTile = tile + gridDim.x;
        if (issuer) {
            if (nextTile < numTiles) {        // prefetch next tile via TDM
                int ns = nextTile * TILE_ELEMS;
                unsigned long long rem = (unsigned long long)(total - ns);
                tdm_load_1d((unsigned)(size_t)&sseg[cur ^ 1][0], seg + ns,
                            TILE_ELEMS, rem);
                tdm_load_1d((unsigned)(size_t)&spid[cur ^ 1][0], pidx + ns,
                            TILE_ELEMS, rem);
                // 4 outstanding; in-order completion => <=2 means current pair done
                __builtin_amdgcn_s_wait_tensorcnt(2);
            } else {
                __builtin_amdgcn_s_wait_tensorcnt(0);
            }
        }
        __syncthreads();                      // publish current tile to all waves

        int start = tile * TILE_ELEMS;
        int base  = threadIdx.x * 8;
        int gbase = start + base;
        int m     = total - gbase; m = m > 8 ? 8 : m;
        if (m > 0) {
            int4 s0 = ((const int4*)&sseg[cur][base])[0];   // ds_load_b128
            int4 s1 = ((const int4*)&sseg[cur][base])[1];
            int4 p0 = ((const int4*)&spid[cur][base])[0];
            int4 p1 = ((const int4*)&spid[cur][base])[1];
            int sg[8] = {s0.x, s0.y, s0.z, s0.w, s1.x, s1.y, s1.z, s1.w};
            float th[8];
            th[0] = theta[p0.x]; th[1] = theta[p0.y];
            th[2] = theta[p0.z]; th[3] = theta[p0.w];
            th[4] = theta[p1.x]; th[5] = theta[p1.y];
            th[6] = theta[p1.z]; th[7] = theta[p1.w];
            accum_chunk(sg, th, m, a_i, b_i, lam, cnt);
        }
        __syncthreads();                      // done reading before buffer reuse
        cur ^= 1;
    }
}
#else
__global__ void __launch_bounds__(256)
k_accum(const int* __restrict__ seg, const int* __restrict__ pidx,
        const float* __restrict__ theta,
        const float* __restrict__ a_i, const float* __restrict__ b_i,
        float* __restrict__ lam, int* __restrict__ cnt, int total) {
    int tid = blockIdx.x * blockDim.x + threadIdx.x;
    int nth = gridDim.x * blockDim.x;
    for (int base = tid * 8; base < total; base += nth * 8) {
        __builtin_prefetch(seg  + base + nth * 8, 0, 1);
        __builtin_prefetch(pidx + base + nth * 8, 0, 1);
        int m = total - base; m = m > 8 ? 8 : m;
        int sg[8]; float th[8];
        if (m == 8) {
            int4 s0 = ((const int4*)(seg  + base))[0];
            int4 s1 = ((const int4*)(seg  + base))[1];
            int4 p0 = ((const int4*)(pidx + base))[0];
            int4 p1 = ((const int4*)(pidx + base))[1];
            sg[0] = s0.x; sg[1] = s0.y; sg[2] = s0.z; sg[3] = s0.w;
            sg[4] = s1.x; sg[5] = s1.y; sg[6] = s1.z; sg[7] = s1.w;
            th[0] = theta[p0.x]; th[1] = theta[p0.y];
            th[2] = theta[p0.z]; th[3] = theta[p0.w];
            th[4] = theta[p1.x]; th[5] = theta[p1.y];
            th[6] = theta[p1.z]; th[7] = theta[p1.w];
        } else {
            for (int j = 0; j < m; ++j) {
                sg[j] = seg[base + j];
                th[j] = theta[pidx[base + j]];
            }
        }
        accum_chunk(sg, th, m, a_i, b_i, lam, cnt);
    }
}
#endif

// ---------------------------------------------------------------------------
// Kernel 5: p = clip(1 - exp(-lam * exp(ts_bias[min(count,10)])), EPS, 1-EPS)
// ---------------------------------------------------------------------------
__global__ void __launch_bounds__(256)
k_final(const float* __restrict__ lam, const int* __restrict__ cnt,
        const float* __restrict__ tsb, float* __restrict__ out, int Bn) {
    __shared__ float s_tsb[MAX_TS + 1];
    if (threadIdx.x <= MAX_TS) s_tsb[threadIdx.x] = tsb[threadIdx.x];
    __syncthreads();

    int i      = blockIdx.x * blockDim.x + threadIdx.x;
    int stride = gridDim.x * blockDim.x;
    for (; i < Bn; i += stride) {
        int c = cnt[i];
        c = c < 0 ? 0 : (c > MAX_TS ? MAX_TS : c);
        float l = lam[i] * __expf(s_tsb[c]);
        float p = 1.0f - __expf(-l);
        out[i] = fminf(fmaxf(p, EPS), 1.0f - EPS);
    }
}

// ---------------------------------------------------------------------------
extern "C" void kernel_launch(void* const* d_in, const int* in_sizes, int n_in,
                              void* d_out, int out_size, void* d_ws, size_t ws_size,
                              hipStream_t stream) {
    const float* theta     = (const float*)d_in[0];
    const float* b         = (const float*)d_in[1];
    const float* log_a     = (const float*)d_in[2];
    const float* tsb       = (const float*)d_in[3];
    const float* dl_scale  = (const float*)d_in[4];
    const float* type_bias = (const float*)d_in[5];
    const float* dl        = (const float*)d_in[6];
    const int*   ttype     = (const int*)d_in[7];
    const int*   qidx      = (const int*)d_in[8];
    const int*   pidx      = (const int*)d_in[9];
    const int*   seg       = (const int*)d_in[10];

    const int NQ    = in_sizes[6];   // 500,000
    const int Bn    = in_sizes[8];   // 4,000,000
    const int total = in_sizes[9];   // 24,000,000

    float* out = (float*)d_out;

    // ws layout: [stats 9f pad64][k01 6f pad64][a_i Bn][b_i Bn][lam Bn][cnt Bn]
    char*  ws    = (char*)d_ws;
    float* stats = (float*)ws;
    float* k01   = (float*)(ws + 64);
    float* a_i   = (float*)(ws + 128);
    float* b_i   = a_i + Bn;
    float* lam   = b_i + Bn;
    int*   cnt   = (int*)(lam + Bn);

    (void)hipMemsetAsync(stats, 0, 64, stream);
    (void)hipMemsetAsync(lam, 0, (size_t)Bn * 2 * sizeof(float), stream);

    {   // 1) per-type dl stats
        int blocks = (NQ + 255) / 256;
        if (blocks > 1024) blocks = 1024;
        k_dl_stats<<<blocks, 256, 0, stream>>>(dl, ttype, NQ, stats);
    }
    // 2) per-type affine coefficients
    k_finalize_stats<<<1, 32, 0, stream>>>(stats, dl_scale, type_bias, k01);

    {   // 3) per-question a_i / b_i
        int blocks = (Bn + 255) / 256;
        if (blocks > 8192) blocks = 8192;
        k_question<<<blocks, 256, 0, stream>>>(qidx, ttype, dl, b, log_a, k01,
                                               a_i, b_i, Bn);
    }
    {   // 4) TDM-staged gather + segmented accumulate
        int numTiles = (total + TILE_ELEMS - 1) / TILE_ELEMS;
        int blocks   = numTiles < 4096 ? numTiles : 4096;
        k_accum<<<blocks, 256, 0, stream>>>(seg, pidx, theta, a_i, b_i,
                                            lam, cnt, total);
    }
    {   // 5) finalize probabilities
        int blocks = (Bn + 255) / 256;
        if (blocks > 8192) blocks = 8192;
        k_final<<<blocks, 256, 0, stream>>>(lam, cnt, tsb, out, Bn);
    }
}